// GetMask_66726611911118
// MI455X (gfx1250) — compile-verified
//
#include <hip/hip_runtime.h>
#include <stdint.h>

// ---------------------------------------------------------------------------
// Problem constants (from the reference): sigma [B=4, C=192, H=512, W=512] f32
// Output: mask [4,1,512,512] f32 followed by values_mask [4,1,512,512] f32.
// Pass 1: WMMA-based channel-mean + threshold + lattice  -> pool bytes (d_ws)
//         (only even rows are ever needed: halves HBM traffic to ~402 MB)
// Pass 2: clamped-shift stencil on pool -> mask & values_mask
// ---------------------------------------------------------------------------

#define BB   4
#define CCH  192
#define HH   512
#define WW   512
#define THRESH 0.18f

typedef __attribute__((ext_vector_type(2))) float v2f;
typedef __attribute__((ext_vector_type(8))) float v8f;

// Pass 1: one wave32 per (b, even row i, 16-column chunk).
//   A (16x4 f32)  : 16 pixels (M) x 4 channels (K)   -- documented lane layout:
//                   lanes 0-15: M=lane, VGPR0=K0, VGPR1=K1
//                   lanes 16-31: M=lane-16, VGPR0=K2, VGPR1=K3
//   B (4x16 f32)  : all ones  => D[m][n] = sum_k A[m][k]  (K-permutation safe)
//   C accumulates over 48 steps -> full 192-channel sum per pixel.
__global__ void __launch_bounds__(256)
pool_pass1_kernel(const float* __restrict__ sigma, unsigned char* __restrict__ pool)
{
    const unsigned tid    = blockIdx.x * blockDim.x + threadIdx.x;
    const unsigned lane   = threadIdx.x & 31u;
    const unsigned waveId = tid >> 5;

    // 32 chunks per row, 256 even rows per batch => 8192 waves per batch
    const unsigned b     = waveId >> 13;
    const unsigned rem   = waveId & 8191u;
    const unsigned row2  = rem >> 5;          // 0..255 (even-row index)
    const unsigned chunk = rem & 31u;
    const int i  = (int)(row2 << 1);          // even row in [0,510]
    const int j0 = (int)(chunk << 4);         // 16 columns per wave
    const int pix = (int)(lane & 15u);        // pixel handled by this lane (M)

    const size_t cstride = (size_t)HH * WW;   // one channel plane
    const float* p = sigma
        + (size_t)b * (size_t)CCH * cstride
        + (size_t)i * WW + (size_t)(j0 + pix)
        + (size_t)((lane >> 4) << 1) * cstride;   // upper lane half: channels +2,+3

    v8f acc = {0.f, 0.f, 0.f, 0.f, 0.f, 0.f, 0.f, 0.f};
    v2f bones;
    bones.x = 1.0f;
    bones.y = 1.0f;

#pragma unroll 4
    for (int c = 0; c < CCH; c += 4) {
        v2f a;
        a.x = __builtin_nontemporal_load(p);            // channel c (+0 / +2)
        a.y = __builtin_nontemporal_load(p + cstride);  // channel c (+1 / +3)
        // v_wmma_f32_16x16x4_f32: D = A x B + C
        acc = __builtin_amdgcn_wmma_f32_16x16x4_f32(
                  false, a, false, bones, (short)0, acc, false, false);
        p += cstride * 4;
    }

    // D layout: VGPR r -> lanes 0-15 hold M=r, lanes 16-31 hold M=r+8
    // (value constant across N).  Lane L selects acc[L&7]:
    //   lanes 0-7   -> pixel L
    //   lanes 16-23 -> pixel L-8
    const unsigned r = lane & 7u;
    float s = acc[0];
    s = (r == 1u) ? acc[1] : s;
    s = (r == 2u) ? acc[2] : s;
    s = (r == 3u) ? acc[3] : s;
    s = (r == 4u) ? acc[4] : s;
    s = (r == 5u) ? acc[5] : s;
    s = (r == 6u) ? acc[6] : s;
    s = (r == 7u) ? acc[7] : s;

    if ((lane & 8u) == 0u) {
        const int m = (lane < 16u) ? (int)lane : (int)lane - 8;  // pixel 0..15
        const int j = j0 + m;
        const float mean = s / 192.0f;
        // lattice: (i%4==0 & j%4==1) | (i%4==2 & j%4==3)  ==  (j&3)==((i&3)|1)
        const int lat = ((j & 3) == ((i & 3) | 1));
        pool[((size_t)b * 256 + row2) * WW + j] =
            (unsigned char)((mean <= THRESH) && lat);
    }
}

// Pass 2: per-pixel clamped-shift stencil on pool bytes.
__global__ void __launch_bounds__(256)
mask_pass2_kernel(const unsigned char* __restrict__ pool, float* __restrict__ out)
{
    const int idx = blockIdx.x * blockDim.x + threadIdx.x;   // 0 .. B*H*W-1
    const int b = idx >> 18;
    const int i = (idx >> 9) & 511;
    const int j = idx & 511;
    const int N = BB * HH * WW;

    const unsigned char* pb = pool + ((size_t)b << 8) * WW;  // [256][512] bytes
    auto P = [&](int ii, int jj) -> int {
        return (ii & 1) ? 0 : (int)pb[(size_t)(ii >> 1) * WW + jj];
    };

    // Faithful clamped-shift semantics (including edge ORs):
    const int bottom = (i == 0)  ? (P(0, j) | P(1, j))
                     : (i < 511) ? P(i + 1, j) : 0;
    const int top    = (i == 0)  ? 0
                     : (i < 511) ? P(i - 1, j) : (P(510, j) | P(511, j));
    const int right  = (j == 0)  ? 0
                     : (j < 511) ? P(i, j - 1) : (P(i, 510) | P(i, 511));
    const int left   = (j == 0)  ? (P(i, 0) | P(i, 1))
                     : (j < 511) ? P(i, j + 1) : 0;

    const float checker = (((i ^ j) & 1) == 0) ? 1.0f : 0.0f;
    out[idx]     = (bottom | top | right) ? 0.0f : checker;
    out[N + idx] = left ? 1.0f : 0.0f;
}

extern "C" void kernel_launch(void* const* d_in, const int* in_sizes, int n_in,
                              void* d_out, int out_size, void* d_ws, size_t ws_size,
                              hipStream_t stream) {
    (void)in_sizes; (void)n_in; (void)out_size; (void)ws_size;

    const float* sigma = (const float*)d_in[0];
    float* out = (float*)d_out;
    unsigned char* pool = (unsigned char*)d_ws;   // needs 4*256*512 = 512 KB

    // Pass 1: 4 batches * 256 even rows * 32 chunks = 32768 waves = 1,048,576 threads
    pool_pass1_kernel<<<4096, 256, 0, stream>>>(sigma, pool);
    // Pass 2: B*H*W = 1,048,576 threads
    mask_pass2_kernel<<<4096, 256, 0, stream>>>(pool, out);
}